// SCAESuite_72911364817215
// MI455X (gfx1250) — compile-verified
//
#include <hip/hip_runtime.h>

// Problem constants (match reference)
#define B_      32
#define KU      64
#define KD      192
#define D_      768
#define SUP     24576
#define C_      256
#define KD_TILE 16
#define NTILES  (KD / KD_TILE)    // 12
#define APAD    4
#define ASTRIDE (D_ + APAD)       // 772 floats: %64 banks == 4, avoids row aliasing

typedef float v2f __attribute__((ext_vector_type(2)));
typedef float v8f __attribute__((ext_vector_type(8)));
typedef int   v4i __attribute__((ext_vector_type(4)));

#if defined(__gfx1250__) && __has_builtin(__builtin_amdgcn_global_load_async_to_lds_b128) && \
    __has_builtin(__builtin_amdgcn_s_wait_asynccnt)
#define HAVE_ASYNC_LDS 1
typedef __attribute__((address_space(1))) v4i* g_v4i;   // global AS pointer
typedef __attribute__((address_space(3))) v4i* l_v4i;   // LDS AS pointer
#else
#define HAVE_ASYNC_LDS 0
#endif

__global__ __launch_bounds__(128) void scae_pruned_contrib(
    const float* __restrict__ up_decoder,    // [D, SUP]      (column gather)
    const float* __restrict__ down_encoder,  // [SDOWN, D]
    const float* __restrict__ up_b_dec,      // [D]
    const float* __restrict__ up_vals,       // [B, KU]
    const int*   __restrict__ connections,   // [SDOWN, C]
    const int*   __restrict__ up_indices,    // [B, KU]
    const int*   __restrict__ down_indices,  // [B, KD]
    float*       __restrict__ out)           // [B, KD]
{
    __shared__ float lA[KD_TILE * ASTRIDE];  // gathered down rows (A tile)
    __shared__ int   lConn[KD_TILE * C_];    // gathered connection rows
    __shared__ float lW[KD_TILE * KU];       // match weights
    __shared__ int   lIdx[KU];
    __shared__ float lVal[KU];
    __shared__ float lOut[KD_TILE];

    const int b    = blockIdx.x;
    const int kt   = blockIdx.y;
    const int tid  = threadIdx.x;   // 0..127
    const int lane = tid & 31;
    const int wave = tid >> 5;      // ku group 0..3

    // ---- stage per-batch indices / values ----
    if (tid < KU) {
        lIdx[tid] = up_indices[b * KU + tid];
        lVal[tid] = up_vals[b * KU + tid];
    }
    if (tid < KD_TILE) lOut[tid] = 0.0f;

    // ---- stage A tile (16 gathered rows x 768 f32) + connection rows ----
#if HAVE_ASYNC_LDS
    for (int i = tid; i < KD_TILE * (D_ / 4); i += 128) {
        int row = i / (D_ / 4);
        int seg = i % (D_ / 4);
        int didx = down_indices[b * KD + kt * KD_TILE + row];
        const float* src = down_encoder + (size_t)didx * D_ + seg * 4;
        __builtin_amdgcn_global_load_async_to_lds_b128(
            (g_v4i)(void*)src, (l_v4i)(void*)&lA[row * ASTRIDE + seg * 4], 0, 0);
    }
    for (int i = tid; i < KD_TILE * (C_ / 4); i += 128) {
        int row = i / (C_ / 4);
        int seg = i % (C_ / 4);
        int didx = down_indices[b * KD + kt * KD_TILE + row];
        const int* src = connections + (size_t)didx * C_ + seg * 4;
        __builtin_amdgcn_global_load_async_to_lds_b128(
            (g_v4i)(void*)src, (l_v4i)(void*)&lConn[row * C_ + seg * 4], 0, 0);
    }
    __builtin_amdgcn_s_wait_asynccnt(0);   // drain this wave's ASYNCcnt pre-barrier
#else
    for (int i = tid; i < KD_TILE * (D_ / 4); i += 128) {
        int row = i / (D_ / 4);
        int seg = i % (D_ / 4);
        int didx = down_indices[b * KD + kt * KD_TILE + row];
        const float4* src = (const float4*)(down_encoder + (size_t)didx * D_);
        *(float4*)(&lA[row * ASTRIDE + seg * 4]) = src[seg];
    }
    for (int i = tid; i < KD_TILE * (C_ / 4); i += 128) {
        int row = i / (C_ / 4);
        int seg = i % (C_ / 4);
        int didx = down_indices[b * KD + kt * KD_TILE + row];
        const int4* src = (const int4*)(connections + (size_t)didx * C_);
        ((int4*)lConn)[row * (C_ / 4) + seg] = src[seg];
    }
#endif
    __syncthreads();

    // ---- match weights: w[row][ku] = count(conn[row]==idx[ku]) * val[ku] ----
    // (-1 pads can never equal a valid index >= 0, so validity is implicit)
    {
        int row = tid >> 3;          // 8 threads per row
        int kb  = (tid & 7) * 8;     // 8 ku each
        int   idx[8];
        float cnt[8];
        #pragma unroll
        for (int j = 0; j < 8; ++j) { idx[j] = lIdx[kb + j]; cnt[j] = 0.0f; }
        const int* cr = &lConn[row * C_];
        for (int c = 0; c < C_; ++c) {
            int cv = cr[c];          // broadcast read across the 8 threads
            #pragma unroll
            for (int j = 0; j < 8; ++j) cnt[j] += (cv == idx[j]) ? 1.0f : 0.0f;
        }
        #pragma unroll
        for (int j = 0; j < 8; ++j) lW[row * KU + kb + j] = cnt[j] * lVal[kb + j];
    }
    __syncthreads();

    // ---- fp32 WMMA GEMM: each wave computes a 16(kd) x 16(ku) tile, K = 768 ----
    // A 16x4 layout: lanes 0-15 -> M=lane, K=d+0/d+1 ; lanes 16-31 -> K=d+2/d+3
    // B 4x16 layout: lanes 0-15 -> N=lane, same K split
    const int nl   = lane & 15;           // M for A, N for B
    const int koff = (lane >> 4) << 1;    // 0 or 2
    const int ku   = wave * 16 + nl;
    const float* colPtr = up_decoder + lIdx[ku];         // element [d][idx] at d*SUP + idx
    const float* aPtr   = &lA[nl * ASTRIDE + koff];

    v8f acc = {};
    #pragma unroll 8
    for (int d = 0; d < D_; d += 4) {
        v2f a = *(const v2f*)(aPtr + d);                 // ds_load_2addr_b64
        v2f bb;
        bb.x = colPtr[(size_t)(d + koff) * SUP];         // strided column gather
        bb.y = colPtr[(size_t)(d + koff + 1) * SUP];
        acc = __builtin_amdgcn_wmma_f32_16x16x4_f32(
                  false, a, false, bb, (short)0, acc, false, false);
    }

    // ---- weighted ku-reduction of the C tile ----
    // C/D layout: lane gives N (=ku), VGPR j gives M: rows j (lanes 0-15) / j+8 (lanes 16-31)
    float partial[8];
    #pragma unroll
    for (int j = 0; j < 8; ++j) {
        int row = j + ((lane >> 4) << 3);
        partial[j] = acc[j] * lW[row * KU + ku];
    }
    #pragma unroll
    for (int j = 0; j < 8; ++j) {        // sum the 16 lanes of each half
        float v = partial[j];
        v += __shfl_xor(v, 1, 32);
        v += __shfl_xor(v, 2, 32);
        v += __shfl_xor(v, 4, 32);
        v += __shfl_xor(v, 8, 32);
        partial[j] = v;
    }
    if (nl == 0) {
        int rbase = (lane >> 4) << 3;
        #pragma unroll
        for (int j = 0; j < 8; ++j)
            atomicAdd(&lOut[rbase + j], partial[j]);     // ds_add_f32
    }

    // ---- bias: dot(down_vec_row, up_b_dec), reusing the staged A tile ----
    {
        int row = tid >> 3;
        int ch  = tid & 7;               // 96 elements each
        float s = 0.0f;
        const float* ar = &lA[row * ASTRIDE + ch * 96];
        const float* br = up_b_dec + ch * 96;
        #pragma unroll 4
        for (int i = 0; i < 96; ++i) s += ar[i] * br[i];
        atomicAdd(&lOut[row], s);
    }
    __syncthreads();

    if (tid < KD_TILE)
        out[b * KD + kt * KD_TILE + tid] = lOut[tid];
}

extern "C" void kernel_launch(void* const* d_in, const int* in_sizes, int n_in,
                              void* d_out, int out_size, void* d_ws, size_t ws_size,
                              hipStream_t stream) {
    const float* up_decoder   = (const float*)d_in[0];
    const float* down_encoder = (const float*)d_in[1];
    const float* up_b_dec     = (const float*)d_in[2];
    const float* up_vals      = (const float*)d_in[3];
    const int*   connections  = (const int*)d_in[4];
    const int*   up_indices   = (const int*)d_in[5];
    const int*   down_indices = (const int*)d_in[6];
    float* out = (float*)d_out;

    dim3 grid(B_, NTILES);
    dim3 block(128);
    scae_pruned_contrib<<<grid, block, 0, stream>>>(
        up_decoder, down_encoder, up_b_dec, up_vals,
        connections, up_indices, down_indices, out);
}